// MyScatter_74526272520549
// MI455X (gfx1250) — compile-verified
//
#include <hip/hip_runtime.h>

typedef __attribute__((ext_vector_type(4))) float v4f;

#define THREADS 256
#define SCAN_T  1024
#define SCAN_V  4                       // elements per thread per chunk
#define CHUNK   (SCAN_T * SCAN_V)       // 4096

// ---------------------------------------------------------------------------
// FAST PATH: bucket-then-gather (counting sort by segment id).
// ---------------------------------------------------------------------------

// Gather-reduce (hot kernel, first in file for disasm visibility).
// One block per segment, blockDim.x == D; thread d owns column d.
// Bucket metadata + rowids are block-uniform -> compiler scalarizes them
// (s_load_b128 of 4 rowids per iteration); each row is one coalesced 256B
// vector transaction, 4 independent loads in flight. No atomics here.
__global__ void __launch_bounds__(64)
gather_reduce_kernel(const float* __restrict__ src,
                     const unsigned* __restrict__ rowids,
                     const unsigned* __restrict__ offs,
                     const unsigned* __restrict__ cnt,
                     float* __restrict__ out,
                     int D) {
    const int n = blockIdx.x;
    const int d = threadIdx.x;

    const unsigned beg = offs[n];
    const unsigned c   = cnt[n];
    const unsigned end = beg + c;

    float acc = 0.0f;
    unsigned i = beg;
    for (; i + 4 <= end; i += 4) {
        unsigned r0 = rowids[i + 0];
        unsigned r1 = rowids[i + 1];
        unsigned r2 = rowids[i + 2];
        unsigned r3 = rowids[i + 3];
        float a0 = __builtin_nontemporal_load(src + (size_t)r0 * D + d);
        float a1 = __builtin_nontemporal_load(src + (size_t)r1 * D + d);
        float a2 = __builtin_nontemporal_load(src + (size_t)r2 * D + d);
        float a3 = __builtin_nontemporal_load(src + (size_t)r3 * D + d);
        acc += a0 + a1 + a2 + a3;
    }
    for (; i < end; ++i) {
        unsigned r = rowids[i];
        acc += __builtin_nontemporal_load(src + (size_t)r * D + d);
    }

    const float denom = (c > 0u) ? (float)c : 1.0f;
    out[(size_t)n * D + d] = acc / denom;
}

// Histogram: one non-returning u32 atomic per row (global_atomic_add_u32).
__global__ void __launch_bounds__(THREADS)
hist_kernel(const int* __restrict__ index, unsigned* __restrict__ cnt, int E) {
    int e = blockIdx.x * blockDim.x + threadIdx.x;
    if (e >= E) return;
    atomicAdd(&cnt[index[e]], 1u);
}

// Wave32 inclusive scan via shfl_up (barrier-free within a wave).
__device__ __forceinline__ unsigned wave_incl_scan(unsigned x, unsigned lane) {
    #pragma unroll
    for (int off = 1; off < 32; off <<= 1) {
        unsigned y = __shfl_up(x, off, 32);
        if (lane >= off) x += y;
    }
    return x;
}

// Single-block chunked exclusive scan: thread-serial(4) -> wave shfl scan ->
// wave0 scans 32 wave totals. 3 barriers per 4096-element chunk (vs ~20 for
// Hillis-Steele). Writes both stable offsets and the working cursor copy.
__global__ void __launch_bounds__(SCAN_T)
scan_kernel(const unsigned* __restrict__ cnt,
            unsigned* __restrict__ offs,
            unsigned* __restrict__ cursor,
            int N) {
    __shared__ unsigned ldsW[32];       // per-wave exclusive offsets
    __shared__ unsigned ldsTot;         // chunk total
    const unsigned tid  = threadIdx.x;
    const unsigned lane = tid & 31u;
    const unsigned wid  = tid >> 5;     // 0..31

    unsigned carry = 0u;                // uniform across block

    for (int base = 0; base < N; base += CHUNK) {
        const int i0 = base + (int)tid * SCAN_V;

        // thread-serial inclusive scan of 4 elements
        unsigned v0 = (i0 + 0 < N) ? cnt[i0 + 0] : 0u;
        unsigned v1 = (i0 + 1 < N) ? cnt[i0 + 1] : 0u;
        unsigned v2 = (i0 + 2 < N) ? cnt[i0 + 2] : 0u;
        unsigned v3 = (i0 + 3 < N) ? cnt[i0 + 3] : 0u;
        unsigned s0 = v0;
        unsigned s1 = s0 + v1;
        unsigned s2 = s1 + v2;
        unsigned s3 = s2 + v3;          // thread total

        // wave-level scan of thread totals (no barriers)
        unsigned wIncl = wave_incl_scan(s3, lane);
        unsigned wExcl = wIncl - s3;
        if (lane == 31u) ldsW[wid] = wIncl;
        __syncthreads();                                    // (1)

        // wave 0 scans the 32 wave totals
        if (wid == 0u) {
            unsigned w  = ldsW[lane];
            unsigned xx = wave_incl_scan(w, lane);
            ldsW[lane]  = xx - w;       // exclusive offset of each wave
            if (lane == 31u) ldsTot = xx;
        }
        __syncthreads();                                    // (2)

        const unsigned tBase = carry + ldsW[wid] + wExcl;   // excl. base of v0
        if (i0 + 0 < N) { unsigned o = tBase;      offs[i0+0] = o; cursor[i0+0] = o; }
        if (i0 + 1 < N) { unsigned o = tBase + s0; offs[i0+1] = o; cursor[i0+1] = o; }
        if (i0 + 2 < N) { unsigned o = tBase + s1; offs[i0+2] = o; cursor[i0+2] = o; }
        if (i0 + 3 < N) { unsigned o = tBase + s2; offs[i0+3] = o; cursor[i0+3] = o; }

        carry += ldsTot;                // uniform update
        __syncthreads();                                    // (3) protect lds reuse
    }
}

// Bucket fill: returning u32 atomic allocates the slot, then plain store.
__global__ void __launch_bounds__(THREADS)
fill_kernel(const int* __restrict__ index,
            unsigned* __restrict__ cursor,
            unsigned* __restrict__ rowids, int E) {
    int e = blockIdx.x * blockDim.x + threadIdx.x;
    if (e >= E) return;
    int seg = index[e];
    unsigned pos = atomicAdd(&cursor[seg], 1u);   // returning atomic
    rowids[pos] = (unsigned)e;
}

__global__ void __launch_bounds__(THREADS)
zero_u32_kernel(unsigned* __restrict__ p, int n) {
    int i = blockIdx.x * blockDim.x + threadIdx.x;
    if (i < n) p[i] = 0u;
}

// ---------------------------------------------------------------------------
// FALLBACK PATH (if ws_size is too small): direct f32 atomic scatter.
// ---------------------------------------------------------------------------

__global__ void __launch_bounds__(THREADS)
scatter_add_kernel(const float* __restrict__ src,
                   const int* __restrict__ index,
                   float* __restrict__ sums,
                   float* __restrict__ counts,
                   int E, int D, int chunkShift, int chunkMask) {
    int gid = blockIdx.x * blockDim.x + threadIdx.x;
    int e   = gid >> chunkShift;
    int c   = gid &  chunkMask;
    if (e >= E) return;

    __builtin_prefetch(src + (size_t)e * D + 4096, 0, 1);   // global_prefetch_b8
    int seg = index[e];

    const v4f* s4 = (const v4f*)(src + (size_t)e * D) + c;
    v4f v = __builtin_nontemporal_load(s4);                 // b128 NT

    float* dst = sums + (size_t)seg * D + (c << 2);
    unsafeAtomicAdd(dst + 0, v.x);                          // global_atomic_add_f32
    unsafeAtomicAdd(dst + 1, v.y);
    unsafeAtomicAdd(dst + 2, v.z);
    unsafeAtomicAdd(dst + 3, v.w);
    if (c == 0) unsafeAtomicAdd(counts + seg, 1.0f);
}

__global__ void __launch_bounds__(THREADS)
scatter_finalize_kernel(float* __restrict__ out,
                        const float* __restrict__ counts,
                        int total, int dShift) {
    int i = blockIdx.x * blockDim.x + threadIdx.x;
    if (i >= total) return;
    float cnt = counts[i >> dShift];
    float denom = cnt > 1.0f ? cnt : 1.0f;
    out[i] = out[i] / denom;
}

__global__ void __launch_bounds__(THREADS)
scatter_zero_kernel(float* __restrict__ out,
                    float* __restrict__ counts,
                    int nOut, int nCounts) {
    int i = blockIdx.x * blockDim.x + threadIdx.x;
    if (i < nOut)    out[i]    = 0.0f;
    if (i < nCounts) counts[i] = 0.0f;
}

static inline int ilog2(int x) { int s = 0; while ((1 << s) < x) ++s; return s; }

extern "C" void kernel_launch(void* const* d_in, const int* in_sizes, int n_in,
                              void* d_out, int out_size, void* d_ws, size_t ws_size,
                              hipStream_t stream) {
    const float* src   = (const float*)d_in[0];
    const int*   index = (const int*)d_in[1];
    const int E = in_sizes[1];              // 800000
    const int D = in_sizes[0] / E;          // 64  (power of two)
    const int N = out_size / D;             // 50000

    float* out = (float*)d_out;

    const size_t need = ((size_t)3 * N + (size_t)E) * sizeof(unsigned);

    if (ws_size >= need) {
        // ---- fast path: bucket-then-gather ----
        unsigned* cnt    = (unsigned*)d_ws;
        unsigned* offs   = cnt    + N;
        unsigned* cursor = offs   + N;
        unsigned* rowids = cursor + N;

        {   // zero counts
            int grid = (N + THREADS - 1) / THREADS;
            zero_u32_kernel<<<grid, THREADS, 0, stream>>>(cnt, N);
        }
        {   // histogram
            int grid = (E + THREADS - 1) / THREADS;
            hist_kernel<<<grid, THREADS, 0, stream>>>(index, cnt, E);
        }
        {   // exclusive scan -> offs, cursor
            scan_kernel<<<1, SCAN_T, 0, stream>>>(cnt, offs, cursor, N);
        }
        {   // bucket fill
            int grid = (E + THREADS - 1) / THREADS;
            fill_kernel<<<grid, THREADS, 0, stream>>>(index, cursor, rowids, E);
        }
        {   // gather-reduce + divide (writes every out element: no pre-zero)
            gather_reduce_kernel<<<N, D, 0, stream>>>(src, rowids, offs, cnt, out, D);
        }
    } else {
        // ---- fallback: direct atomic scatter ----
        const int chunks     = D >> 2;
        const int chunkShift = ilog2(chunks);
        const int chunkMask  = chunks - 1;
        const int dShift     = ilog2(D);
        float* counts = (float*)d_ws;       // N floats

        {
            int grid = (out_size + THREADS - 1) / THREADS;
            scatter_zero_kernel<<<grid, THREADS, 0, stream>>>(out, counts, out_size, N);
        }
        {
            long long total = (long long)E * chunks;
            int grid = (int)((total + THREADS - 1) / THREADS);
            scatter_add_kernel<<<grid, THREADS, 0, stream>>>(src, index, out, counts,
                                                             E, D, chunkShift, chunkMask);
        }
        {
            int grid = (out_size + THREADS - 1) / THREADS;
            scatter_finalize_kernel<<<grid, THREADS, 0, stream>>>(out, counts, out_size, dShift);
        }
    }
}